// BaselineModel_42348377538683
// MI455X (gfx1250) — compile-verified
//
#include <hip/hip_runtime.h>
#include <stdint.h>

// Segment-mean embedding pooling for MI455X (gfx1250).
//
// Roofline: ~600 MB gathered + 30 MB stored ≈ 630 MB -> ~27 us floor at
// 23.3 TB/s HBM; 1.5e8 FLOPs is noise (0.25 FLOP/byte), so no WMMA — the
// CDNA5 win is the async global->LDS DMA path (ASYNCcnt) + wide b128 ops.
//
// One wave32 per segment: no barriers, each wave owns its LDS slice, so
// s_wait_asynccnt is the only sync. Wave-uniform scalars are pinned with
// readfirstlane so the segment binary search runs on SALU/SMEM instead of
// 38 serialized per-lane VMEM round-trips.

#define EMB_DIM        300
#define F4_PER_ROW     75                  // 300 floats = 75 float4 (1200 B)
#define TAIL_LANES     (F4_PER_ROW - 64)   // 11 lanes active in chunk 2
#define WAVES_PER_BLK  8
#define SLOTS          4                   // rows staged per wave per round
#define THREADS        (WAVES_PER_BLK * 32)

typedef float f32x4 __attribute__((ext_vector_type(4)));

__global__ __launch_bounds__(THREADS)
void seg_mean_pool_kernel(const float* __restrict__ emb,
                          const int*   __restrict__ ids,
                          const int*   __restrict__ segs,
                          float*       __restrict__ out,
                          int total_words, int num_segments)
{
    // 8 waves * 4 slots * 75 float4 = 38400 B static LDS
    __shared__ f32x4 stage[WAVES_PER_BLK][SLOTS][F4_PER_ROW];

    const int lane = (int)(threadIdx.x & 31u);
    const int wid  = (int)(threadIdx.x >> 5);
    // seg is wave-uniform; make that visible to the compiler (SALU search).
    const int seg  = __builtin_amdgcn_readfirstlane((int)blockIdx.x * WAVES_PER_BLK + wid);
    if (seg >= num_segments) return;

    // Scalar lower_bound over sorted segment_ids (s_load_b32 / KMcnt path;
    // the 2 MB segs array is L2-resident after first touch).
    int lo = 0, hi = total_words;
    while (lo < hi) { int mid = (lo + hi) >> 1; if (segs[mid] < seg) lo = mid + 1; else hi = mid; }
    const int start = __builtin_amdgcn_readfirstlane(lo);
    lo = start; hi = total_words;
    while (lo < hi) { int mid = (lo + hi) >> 1; if (segs[mid] < seg + 1) lo = mid + 1; else hi = mid; }
    const int end   = __builtin_amdgcn_readfirstlane(lo);
    const int count = end - start;

    f32x4 acc0 = {0.f, 0.f, 0.f, 0.f};
    f32x4 acc1 = {0.f, 0.f, 0.f, 0.f};
    f32x4 acc2 = {0.f, 0.f, 0.f, 0.f};

    // Flat addr low 32 bits == LDS byte offset (LDS aperture truncation).
    const unsigned lds_wave_base = (unsigned)(uintptr_t)(&stage[wid][0][0]);
    const unsigned lane_off0 = (unsigned)lane * 16u;
    const unsigned lane_off1 = lane_off0 + 32u * 16u;
    const unsigned lane_off2 = lane_off0 + 64u * 16u;

    for (int base = start; base < end; base += SLOTS) {
        const int nw = min(SLOTS, end - base);

        // WAR guard: previous round's ds_load_b128 reads of these slots must
        // drain before the async engine overwrites them.
        asm volatile("s_wait_dscnt 0x0" ::: "memory");

        // ---- stage nw gathered rows into LDS via async DMA (3 x b128/row) ----
        for (int s = 0; s < nw; ++s) {
            const int row = __builtin_amdgcn_readfirstlane(ids[base + s]);  // uniform
            const uint64_t gsrc = (uint64_t)(uintptr_t)emb
                                + (uint64_t)(unsigned)row * (uint64_t)(EMB_DIM * 4);
            const unsigned lbase = lds_wave_base + (unsigned)s * (F4_PER_ROW * 16u);
            {
                uint64_t g = gsrc + (uint64_t)lane_off0;
                unsigned l = lbase + lane_off0;
                asm volatile("global_load_async_to_lds_b128 %0, %1, off"
                             :: "v"(l), "v"(g) : "memory");
            }
            {
                uint64_t g = gsrc + (uint64_t)lane_off1;
                unsigned l = lbase + lane_off1;
                asm volatile("global_load_async_to_lds_b128 %0, %1, off"
                             :: "v"(l), "v"(g) : "memory");
            }
            if (lane < TAIL_LANES) {
                uint64_t g = gsrc + (uint64_t)lane_off2;
                unsigned l = lbase + lane_off2;
                asm volatile("global_load_async_to_lds_b128 %0, %1, off"
                             :: "v"(l), "v"(g) : "memory");
            }
        }

        // Wait for this wave's async copies (ASYNCcnt, in-order completion).
        asm volatile("s_wait_asynccnt 0x0" ::: "memory");

        // ---- consume staged rows from LDS (ds_load_b128) ----
        for (int s = 0; s < nw; ++s) {
            const f32x4* r = &stage[wid][s][0];
            acc0 += r[lane];
            acc1 += r[32 + lane];
            if (lane < TAIL_LANES) acc2 += r[64 + lane];
        }
    }

    const float inv = 1.0f / (float)count;   // mean divisor == segment length
    acc0 *= inv; acc1 *= inv; acc2 *= inv;

    // Output is write-once/stream: NT stores keep it from evicting the
    // embedding table's L2 lines (table has real reuse; output has none).
    f32x4* o = (f32x4*)(out + (size_t)seg * EMB_DIM);   // 1200 B rows, 16B aligned
    __builtin_nontemporal_store(acc0, o + lane);
    __builtin_nontemporal_store(acc1, o + 32 + lane);
    if (lane < TAIL_LANES) __builtin_nontemporal_store(acc2, o + 64 + lane);
}

extern "C" void kernel_launch(void* const* d_in, const int* in_sizes, int n_in,
                              void* d_out, int out_size, void* d_ws, size_t ws_size,
                              hipStream_t stream) {
    const float* word_emb    = (const float*)d_in[0];
    const int*   word_ids    = (const int*)  d_in[1];
    const int*   segment_ids = (const int*)  d_in[2];
    // d_in[3] (num_segments) lives on device; derive the same value from out_size.
    float* out = (float*)d_out;

    const int total_words  = in_sizes[1];
    const int num_segments = out_size / EMB_DIM;

    const int grid = (num_segments + WAVES_PER_BLK - 1) / WAVES_PER_BLK;
    seg_mean_pool_kernel<<<grid, THREADS, 0, stream>>>(
        word_emb, word_ids, segment_ids, out, total_words, num_segments);
}